// YatPerformerCausalAttention_64793876627787
// MI455X (gfx1250) — compile-verified
//
#include <hip/hip_runtime.h>
#include <hip/hip_bf16.h>

// ---------------------------------------------------------------------------
// CDNA5 (gfx1250) Yat-Performer causal attention, all-f32 via V_WMMA_F32_16X16X4_F32
// ---------------------------------------------------------------------------

typedef float v2f __attribute__((ext_vector_type(2)));
typedef float v8f __attribute__((ext_vector_type(8)));

static __device__ __forceinline__ v8f wmma4(v2f a, v2f b, v8f c) {
  // D(16x16,f32) = A(16x4,f32) * B(4x16,f32) + C
  return __builtin_amdgcn_wmma_f32_16x16x4_f32(false, a, false, b, (short)0, c,
                                               false, false);
}

static __device__ __forceinline__ v8f vzero8() {
  v8f z;
#pragma unroll
  for (int i = 0; i < 8; ++i) z[i] = 0.0f;
  return z;
}

// ---------------------------------------------------------------------------
// Generic tiled GEMM:  C[M,N] = A[M,K] @ B[K,N] + bias[N]
// mode 0: row-major store to dst0
// mode 1: qkv head-scatter: col -> (part p, head h, dim d); store to
//         dst{p}[((b*16+h)*4096 + l)*64 + d]   (q/k/v head layout)
// ---------------------------------------------------------------------------
#define GBM 128
#define GBN 128
#define GBK 16
#define ASTR 20    // As row stride (floats): 16B-aligned float4 stores, distinct banks
#define BSTR 144   // Bs row stride (floats): disjoint bank groups for the 4 K-rows

__global__ __launch_bounds__(256) void gemm_wmma_f32(
    const float* __restrict__ A, const float* __restrict__ B,
    const float* __restrict__ bias, int M, int N, int K, int mode,
    float* __restrict__ dst0, float* __restrict__ dst1, float* __restrict__ dst2) {
  __shared__ float As[GBM * ASTR];
  __shared__ float Bs[GBK * BSTR];

  const int tid  = threadIdx.x;
  const int lane = tid & 31;
  const int wave = tid >> 5;
  const int waveM = wave & 3;   // 4 waves along M
  const int waveN = wave >> 2;  // 2 waves along N
  const int ln = lane & 15;            // column/row-in-tile index
  const int kp = (lane >> 4) << 1;     // K-pair select (0 or 2)

  const int bm = blockIdx.y * GBM;
  const int bn = blockIdx.x * GBN;

  v8f acc[2][4];
#pragma unroll
  for (int mt = 0; mt < 2; ++mt)
#pragma unroll
    for (int nt = 0; nt < 4; ++nt) acc[mt][nt] = vzero8();

  for (int k0 = 0; k0 < K; k0 += GBK) {
    // ---- global -> LDS (A: 128x16, B: 16x128), float4, fully coalesced ----
#pragma unroll
    for (int r = 0; r < 2; ++r) {
      int id  = tid * 2 + r;                 // 0..511 float4 slots
      int row = id >> 2, kq = (id & 3) << 2;
      float4 va = *(const float4*)(A + (size_t)(bm + row) * K + k0 + kq);
      *(float4*)(&As[row * ASTR + kq]) = va;
    }
#pragma unroll
    for (int r = 0; r < 2; ++r) {
      int id = tid * 2 + r;
      int kr = id >> 5, cq = (id & 31) << 2;
      float4 vb = *(const float4*)(B + (size_t)(k0 + kr) * N + bn + cq);
      *(float4*)(&Bs[kr * BSTR + cq]) = vb;
    }
    __syncthreads();

#pragma unroll
    for (int kk = 0; kk < GBK; kk += 4) {
      v2f bfrag[4];
#pragma unroll
      for (int nt = 0; nt < 4; ++nt) {
        int col = waveN * 64 + nt * 16 + ln;
        bfrag[nt].x = Bs[(kk + kp) * BSTR + col];
        bfrag[nt].y = Bs[(kk + kp + 1) * BSTR + col];
      }
#pragma unroll
      for (int mt = 0; mt < 2; ++mt) {
        int row = waveM * 32 + mt * 16 + ln;
        v2f afrag = *(const v2f*)(&As[row * ASTR + kk + kp]);  // 8B aligned
#pragma unroll
        for (int nt = 0; nt < 4; ++nt)
          acc[mt][nt] = wmma4(afrag, bfrag[nt], acc[mt][nt]);
      }
    }
    __syncthreads();
  }

  // ---- epilogue ----
#pragma unroll
  for (int mt = 0; mt < 2; ++mt) {
#pragma unroll
    for (int nt = 0; nt < 4; ++nt) {
      int colg = bn + waveN * 64 + nt * 16 + ln;
      float bv = bias[colg];
#pragma unroll
      for (int i = 0; i < 8; ++i) {
        int rowg = bm + waveM * 32 + mt * 16 + i + ((lane >> 4) << 3);
        float v = acc[mt][nt][i] + bv;
        if (mode == 0) {
          dst0[(size_t)rowg * N + colg] = v;
        } else {
          int p = colg >> 10, e = colg & 1023;
          int h = e >> 6, d = e & 63;
          int b = rowg >> 12, l = rowg & 4095;
          float* dp = (p == 0) ? dst0 : ((p == 1) ? dst1 : dst2);
          dp[(((size_t)(b * 16 + h) * 4096 + l) << 6) + d] = v;
        }
      }
    }
  }
}

// ---------------------------------------------------------------------------
// Feature kernel: in-place transform of q/k head buffers [BH][L][64]:
//   xn = x/||x||;  proj[f] = sum_d xn[d]*Om[d][f]  (f = r*32+m)
//   fused = (proj^2/sqrtM) * exp(clip(proj*sqrt(2s_r)-s_r,+-20))/sqrtM * sqrt(w_r)
// D == F == 64, so read/write the same 64 floats per token (in-place safe:
// all lanes' loads complete before any store, single wave owns a token).
// ---------------------------------------------------------------------------
__global__ __launch_bounds__(256) void feat_kernel(
    float* __restrict__ qbuf, float* __restrict__ kbuf,
    const float* __restrict__ omega, const float* __restrict__ qn,
    const float* __restrict__ qw) {
  __shared__ float Om[64 * 65];   // stride 65 -> conflict-free column reads
  __shared__ float xb[8][64];

  const int tid = threadIdx.x, lane = tid & 31, wave = tid >> 5;
  const int bh = blockIdx.y;      // b*16+h
  const int h = bh & 15;
  const int l0 = blockIdx.x * 64;
  float* buf = blockIdx.z ? kbuf : qbuf;

  for (int idx = tid; idx < 4096; idx += 256) {
    int d = idx >> 6, f = idx & 63;
    int r = f >> 5, m = f & 31;
    Om[d * 65 + f] = omega[(((size_t)r * 16 + h) * 64 + d) * 32 + m];
  }
  const float s0 = qn[0], s1 = qn[1], w0 = qw[0], w1 = qw[1];
  __syncthreads();

  const float inv_sqrt_m = 0.17677669529663687f;  // 1/sqrt(32)
  const float q0 = sqrtf(2.0f * fmaxf(s0, 0.0f));
  const float q1 = sqrtf(2.0f * fmaxf(s1, 0.0f));
  const float sw0 = sqrtf(fmaxf(w0, 0.0f));
  const float sw1 = sqrtf(fmaxf(w1, 0.0f));

  for (int t = 0; t < 8; ++t) {
    int l = l0 + wave * 8 + t;
    float* vp = buf + (((size_t)bh * 4096 + l) << 6);
    float v0 = vp[2 * lane], v1 = vp[2 * lane + 1];
    float ss = v0 * v0 + v1 * v1;
#pragma unroll
    for (int off = 16; off; off >>= 1) ss += __shfl_xor(ss, off, 32);
    float inv = rsqrtf(ss);
    xb[wave][2 * lane] = v0 * inv;
    xb[wave][2 * lane + 1] = v1 * inv;
    __builtin_amdgcn_wave_barrier();  // LDS same-wave in-order; stop reordering

    float p0 = 0.0f, p1 = 0.0f;
#pragma unroll 8
    for (int d = 0; d < 64; ++d) {
      float xv = xb[wave][d];                 // broadcast read
      p0 = fmaf(xv, Om[d * 65 + lane], p0);
      p1 = fmaf(xv, Om[d * 65 + 32 + lane], p1);
    }
    float a0 = fminf(fmaxf(fmaf(p0, q0, -s0), -20.0f), 20.0f);
    float a1 = fminf(fmaxf(fmaf(p1, q1, -s1), -20.0f), 20.0f);
    float o0 = (p0 * p0 * inv_sqrt_m) * (expf(a0) * inv_sqrt_m) * sw0;
    float o1 = (p1 * p1 * inv_sqrt_m) * (expf(a1) * inv_sqrt_m) * sw1;
    vp[lane] = o0;          // f = lane      (r=0)
    vp[32 + lane] = o1;     // f = 32+lane   (r=1)
    __builtin_amdgcn_wave_barrier();
  }
}

// ---------------------------------------------------------------------------
// Chunked causal linear-attention scan. One block per (b,h). Chunk T=32.
//   S = Qf Kf^T (inclusive causal mask);  O = Qf@KV_prev + S_m@V
//   denom = rowsum(S_m) + Qf . kstate_prev + 1e-6
//   KV += Kf^T @ V ;  kstate += colsum(Kf)
// All 4 products through V_WMMA_F32_16X16X4_F32; state KV(64x64) in LDS.
// ---------------------------------------------------------------------------
#define FS 68   // padded row stride for 64-wide tiles (even -> 8B-aligned v2f)

__global__ __launch_bounds__(256) void scan_kernel(
    const float* __restrict__ qf, const float* __restrict__ kf,
    const float* __restrict__ vv, float* __restrict__ attn) {
  __shared__ float Qs[32 * FS], Ks[32 * FS], Vs[32 * FS];
  __shared__ float Ss[32 * 36];
  __shared__ float KV[64 * FS];
  __shared__ float kstate[64], kcol[64], denom[32];

  const int tid = threadIdx.x, lane = tid & 31, wave = tid >> 5;
  const int ln = lane & 15;
  const int kp = (lane >> 4) << 1;
  const int bh = blockIdx.x;
  const int h = bh & 15, b = bh >> 4;

  const float* qb = qf + ((size_t)bh * 4096 << 6);
  const float* kb = kf + ((size_t)bh * 4096 << 6);
  const float* vb = vv + ((size_t)bh * 4096 << 6);

  for (int i = tid; i < 64 * FS; i += 256) KV[i] = 0.0f;
  if (tid < 64) kstate[tid] = 0.0f;
  __syncthreads();

  // This wave's O tile (32x64 -> 8 tiles of 16x16)
  const int om = (wave & 1) << 4, on = (wave >> 1) << 4;

  for (int c = 0; c < 128; ++c) {
    const int l0 = c * 32;
    // ---- load chunk (each source row is 64 contiguous floats) ----
#pragma unroll
    for (int r = 0; r < 2; ++r) {
      int id = tid * 2 + r;              // 0..511 float4 slots
      int t = id >> 4, q = (id & 15) << 2;
      *(float4*)(&Qs[t * FS + q]) = *(const float4*)(qb + ((size_t)(l0 + t) << 6) + q);
      *(float4*)(&Ks[t * FS + q]) = *(const float4*)(kb + ((size_t)(l0 + t) << 6) + q);
      *(float4*)(&Vs[t * FS + q]) = *(const float4*)(vb + ((size_t)(l0 + t) << 6) + q);
    }
    __syncthreads();  // A

    // ---- O_inter = Qf @ KV_prev  (K = F = 64) ----
    v8f oacc = vzero8();
#pragma unroll
    for (int kk = 0; kk < 64; kk += 4) {
      v2f a = *(const v2f*)(&Qs[(om + ln) * FS + kk + kp]);
      v2f bf;
      bf.x = KV[(kk + kp) * FS + on + ln];
      bf.y = KV[(kk + kp + 1) * FS + on + ln];
      oacc = wmma4(a, bf, oacc);
    }

    // ---- S = Qf @ Kf^T on waves 0..3, masked (inclusive causal) ----
    if (wave < 4) {
      int sm = (wave & 1) << 4, sn = (wave >> 1) << 4;
      v8f sacc = vzero8();
#pragma unroll
      for (int kk = 0; kk < 64; kk += 4) {
        v2f a = *(const v2f*)(&Qs[(sm + ln) * FS + kk + kp]);
        v2f bf = *(const v2f*)(&Ks[(sn + ln) * FS + kk + kp]);  // B[k][n]=Kf[n][k]
        sacc = wmma4(a, bf, sacc);
      }
#pragma unroll
      for (int i = 0; i < 8; ++i) {
        int tq = sm + i + ((lane >> 4) << 3);
        int tk = sn + ln;
        Ss[tq * 36 + tk] = (tk <= tq) ? sacc[i] : 0.0f;
      }
    }
    __syncthreads();  // D

    // ---- denominators + column sums of Kf (disjoint thread groups) ----
    if (tid < 32) {
      float rs = 0.0f;
#pragma unroll 8
      for (int j = 0; j < 32; ++j) rs += Ss[tid * 36 + j];
      float dq = 0.0f;
#pragma unroll 8
      for (int f = 0; f < 64; ++f) dq = fmaf(Qs[tid * FS + f], kstate[f], dq);
      denom[tid] = rs + dq + 1e-6f;
    } else if (tid >= 64 && tid < 128) {
      int f = tid - 64;
      float s = 0.0f;
#pragma unroll 8
      for (int t = 0; t < 32; ++t) s += Ks[t * FS + f];
      kcol[f] = s;
    }

    // ---- O += S_masked @ V  (K = T = 32) ----
#pragma unroll
    for (int kk = 0; kk < 32; kk += 4) {
      v2f a = *(const v2f*)(&Ss[(om + ln) * 36 + kk + kp]);
      v2f bf;
      bf.x = Vs[(kk + kp) * FS + on + ln];
      bf.y = Vs[(kk + kp + 1) * FS + on + ln];
      oacc = wmma4(a, bf, oacc);
    }
    __syncthreads();  // G  (denom/kcol ready; KV B-reads finished)

    // ---- write output: out = O / denom ----
#pragma unroll
    for (int i = 0; i < 8; ++i) {
      int t = om + i + ((lane >> 4) << 3);
      int d = on + ln;
      float v = oacc[i] / denom[t];
      attn[((size_t)b * 4096 + l0 + t) * 1024 + h * 64 + d] = v;
    }

    // ---- KV += Kf^T @ V  (2 of 16 tiles per wave, K = 32) ----
#pragma unroll
    for (int j = 0; j < 2; ++j) {
      int ti = wave * 2 + j;
      int tm = (ti & 3) << 4, tn = (ti >> 2) << 4;
      v8f cacc;
#pragma unroll
      for (int i = 0; i < 8; ++i)
        cacc[i] = KV[(tm + i + ((lane >> 4) << 3)) * FS + tn + ln];
#pragma unroll
      for (int kk = 0; kk < 32; kk += 4) {
        v2f a;  // A[m=f][k=t] = Kf[t][f]
        a.x = Ks[(kk + kp) * FS + tm + ln];
        a.y = Ks[(kk + kp + 1) * FS + tm + ln];
        v2f bf;
        bf.x = Vs[(kk + kp) * FS + tn + ln];
        bf.y = Vs[(kk + kp + 1) * FS + tn + ln];
        cacc = wmma4(a, bf, cacc);
      }
#pragma unroll
      for (int i = 0; i < 8; ++i)
        KV[(tm + i + ((lane >> 4) << 3)) * FS + tn + ln] = cacc[i];
    }
    if (tid < 64) kstate[tid] += kcol[tid];
    __syncthreads();  // I (state committed before next chunk)
  }
}

// ---------------------------------------------------------------------------
// Host launcher
// ---------------------------------------------------------------------------
extern "C" void kernel_launch(void* const* d_in, const int* in_sizes, int n_in,
                              void* d_out, int out_size, void* d_ws, size_t ws_size,
                              hipStream_t stream) {
  const float* x        = (const float*)d_in[0];  // [2,4096,1024]
  const float* W_qkv    = (const float*)d_in[1];  // [1024,3072]
  const float* b_qkv    = (const float*)d_in[2];  // [3072]
  const float* W_out    = (const float*)d_in[3];  // [1024,1024]
  const float* b_out    = (const float*)d_in[4];  // [1024]
  const float* omega    = (const float*)d_in[5];  // [2,16,64,32]
  const float* qnodes   = (const float*)d_in[6];  // [2]
  const float* qweights = (const float*)d_in[7];  // [2]

  float* ws = (float*)d_ws;
  const size_t HBUF = (size_t)2 * 16 * 4096 * 64;  // 8,388,608 floats (32 MB)
  float* wq = ws;              // q heads, then q_feat (in place)
  float* wk = wq + HBUF;       // k heads, then k_feat (in place)
  float* wv = wk + HBUF;       // v heads
  float* wa = wv + HBUF;       // attention output [8192,1024]

  // 1) QKV GEMM, scatter to head layout
  gemm_wmma_f32<<<dim3(3072 / GBN, 8192 / GBM), 256, 0, stream>>>(
      x, W_qkv, b_qkv, 8192, 3072, 1024, /*mode=*/1, wq, wk, wv);

  // 2) random-feature transform of q and k (in place)
  feat_kernel<<<dim3(4096 / 64, 32, 2), 256, 0, stream>>>(wq, wk, omega, qnodes,
                                                          qweights);

  // 3) chunked causal linear-attention scan
  scan_kernel<<<dim3(32), 256, 0, stream>>>(wq, wk, wv, wa);

  // 4) output projection
  gemm_wmma_f32<<<dim3(1024 / GBN, 8192 / GBM), 256, 0, stream>>>(
      wa, W_out, b_out, 8192, 1024, 1024, /*mode=*/0, (float*)d_out, nullptr,
      nullptr);
}